// _IPALayer_15161234555331
// MI455X (gfx1250) — compile-verified
//
#include <hip/hip_runtime.h>
#include <hip/hip_bf16.h>
#include <math.h>

typedef __attribute__((ext_vector_type(2))) float v2f;
typedef __attribute__((ext_vector_type(8))) float v8f;

#define D_SINGLE 384
#define D_PAIR   128
#define N_HEADS  12
#define HEAD_DIM 32
#define SEQ_L    1024

static __device__ __forceinline__ v8f wmma4(v2f a, v2f b, v8f c) {
    // D = A(16x4 f32) * B(4x16 f32) + C(16x16 f32)
    return __builtin_amdgcn_wmma_f32_16x16x4_f32(
        /*neg_a=*/false, a, /*neg_b=*/false, b,
        /*c_mod=*/(short)0, c, /*reuse_a=*/false, /*reuse_b=*/false);
}

// ---------------------------------------------------------------------------
// LayerNorm over last dim (384), one block (128 threads) per row.
// ---------------------------------------------------------------------------
__global__ void ln_kernel(const float* __restrict__ x,
                          const float* __restrict__ g,
                          const float* __restrict__ b,
                          float* __restrict__ y) {
    __shared__ float red[2][128];
    const int row = blockIdx.x;
    const int t = threadIdx.x;
    const float* xr = x + (size_t)row * D_SINGLE;
    float v0 = xr[t], v1 = xr[t + 128], v2 = xr[t + 256];
    red[0][t] = v0 + v1 + v2;
    red[1][t] = v0 * v0 + v1 * v1 + v2 * v2;
    __syncthreads();
    for (int off = 64; off > 0; off >>= 1) {
        if (t < off) {
            red[0][t] += red[0][t + off];
            red[1][t] += red[1][t + off];
        }
        __syncthreads();
    }
    const float mean = red[0][0] * (1.0f / D_SINGLE);
    const float var  = red[1][0] * (1.0f / D_SINGLE) - mean * mean;
    const float inv  = rsqrtf(var + 1e-5f);
    float* yr = y + (size_t)row * D_SINGLE;
    yr[t]       = (v0 - mean) * inv * g[t]       + b[t];
    yr[t + 128] = (v1 - mean) * inv * g[t + 128] + b[t + 128];
    yr[t + 256] = (v2 - mean) * inv * g[t + 256] + b[t + 256];
}

// ---------------------------------------------------------------------------
// Generic fp32 WMMA GEMM: C[M,N] = act(A[M,K] @ B[K,N] + bias) + resid
// Block = 128 threads = 4 waves; wave computes one 16x16 tile.
// Grid: (N/64, M/16). act: 0 = none, 1 = exact GELU.
// ---------------------------------------------------------------------------
__global__ void gemm_kernel(const float* __restrict__ A,
                            const float* __restrict__ B,
                            const float* __restrict__ bias,
                            const float* __restrict__ resid,
                            float* __restrict__ C,
                            int M, int N, int K, int act) {
    const int wave = threadIdx.x >> 5;
    const int lane = threadIdx.x & 31;
    const int lm   = lane & 15;          // row (A) / col (B,C) within tile
    const int half = lane >> 4;
    const int klo  = half * 2;           // K-slot: lanes 0-15 -> k+0,k+1 ; 16-31 -> k+2,k+3
    const int mbase = blockIdx.y * 16;
    const int nbase = blockIdx.x * 64 + wave * 16;

    const float* arow = A + (size_t)(mbase + lm) * K + klo;
    const float* bp   = B + (size_t)klo * N + nbase + lm;

    v8f acc = {};
#pragma unroll 8
    for (int k0 = 0; k0 < K; k0 += 4) {
        v2f a = *(const v2f*)(arow + k0);
        v2f bv;
        bv.x = bp[(size_t)k0 * N];
        bv.y = bp[(size_t)(k0 + 1) * N];
        acc = wmma4(a, bv, acc);
    }

#pragma unroll
    for (int r = 0; r < 8; ++r) {
        const int row = mbase + r + 8 * half;
        const int col = nbase + lm;
        float x = acc[r];
        if (bias) x += bias[col];
        if (act == 1) x = 0.5f * x * (1.0f + erff(x * 0.70710678118654752f));
        if (resid) x += resid[(size_t)row * N + col];
        C[(size_t)row * N + col] = x;
    }
}

// ---------------------------------------------------------------------------
// Pair bias: bias[q][k][h] = sum_p pair[q,k,p] * Wb[p,h]   (h < 12, N padded 16)
// Unit = (q, k-tile of 16). 4 waves/block, one unit per wave.
// Pair tile (16 rows x 128 ch = 8KB, contiguous) staged via coalesced float4.
// ---------------------------------------------------------------------------
__global__ void pair_bias_kernel(const float* __restrict__ pair,
                                 const float* __restrict__ Wb,
                                 float* __restrict__ bias) {
    __shared__ alignas(16) float sWb[D_PAIR * N_HEADS];    // 6 KB
    __shared__ alignas(16) float sTile[4][16 * D_PAIR];    // 4 x 8 KB

    for (int i = threadIdx.x; i < D_PAIR * N_HEADS; i += 128) sWb[i] = Wb[i];
    __syncthreads();

    const int wave = threadIdx.x >> 5;
    const int lane = threadIdx.x & 31;
    const int u  = blockIdx.x * 4 + wave;   // 1024 q * 64 ktiles = 65536 units
    const int qr = u >> 6;
    const int kbase = (u & 63) * 16;

    // stage 2048 contiguous floats (rows kbase..kbase+15 of q-row qr)
    const float4* src = (const float4*)(pair + ((size_t)qr * SEQ_L + kbase) * D_PAIR);
    float4* dst = (float4*)sTile[wave];
#pragma unroll
    for (int i = 0; i < 16; ++i) dst[lane + i * 32] = src[lane + i * 32];
    __builtin_amdgcn_wave_barrier();

    const int lm   = lane & 15;
    const int half = lane >> 4;
    const int klo  = half * 2;
    const float* tile = sTile[wave];

    v8f acc = {};
#pragma unroll
    for (int p0 = 0; p0 < D_PAIR; p0 += 4) {
        v2f a = *(const v2f*)(tile + lm * D_PAIR + p0 + klo);   // A: M=k_local, K=p
        v2f bv;
        const bool hv = (lm < N_HEADS);
        bv.x = hv ? sWb[(p0 + klo)     * N_HEADS + lm] : 0.0f;  // B: K=p, N=h
        bv.y = hv ? sWb[(p0 + klo + 1) * N_HEADS + lm] : 0.0f;
        acc = wmma4(a, bv, acc);
    }

    if (lm < N_HEADS) {
#pragma unroll
        for (int r = 0; r < 8; ++r) {
            const int kl = r + 8 * half;
            bias[((size_t)qr * SEQ_L + kbase + kl) * N_HEADS + lm] = acc[r];
        }
    }
}

// ---------------------------------------------------------------------------
// Flash-style attention. One wave per (head, q-tile of 16 rows).
// logits = (Q*scale)K^T + bias ; online softmax (m,l in LDS) ; acc += P@V.
// P round-trips through LDS to remap C/D layout -> A layout.
// ---------------------------------------------------------------------------
__global__ void attn_kernel(const float* __restrict__ qm,
                            const float* __restrict__ km,
                            const float* __restrict__ vm,
                            const float* __restrict__ bias,
                            const unsigned char* __restrict__ mask,
                            float* __restrict__ att) {
    __shared__ alignas(16) float lds[4 * 320];   // per wave: P[256] + m[16] + l[16]
    const int wave = threadIdx.x >> 5;
    const int lane = threadIdx.x & 31;
    const int u  = blockIdx.x * 4 + wave;        // 12 heads * 64 q-tiles = 768 units
    const int h  = u % N_HEADS;
    const int qbase = (u / N_HEADS) * 16;
    const int lm   = lane & 15;
    const int half = lane >> 4;
    const int klo  = half * 2;

    float* P  = lds + wave * 320;
    float* Lm = P + 256;
    float* Ll = Lm + 16;
    if (lane < 16) { Lm[lane] = -1e30f; Ll[lane] = 0.0f; }
    __builtin_amdgcn_wave_barrier();

    const float scale = 0.17677669529663687f;    // 1/sqrt(32)

    // Preload Q tile in A-layout, scale folded in. (16 rows x 32 dims)
    v2f aq[8];
#pragma unroll
    for (int c = 0; c < 8; ++c) {
        const float* p = qm + (size_t)(qbase + lm) * D_SINGLE + h * HEAD_DIM + c * 4 + klo;
        aq[c].x = p[0] * scale;
        aq[c].y = p[1] * scale;
    }

    v8f acc0 = {}, acc1 = {};                    // d = 0..15 and 16..31

    for (int kt = 0; kt < 64; ++kt) {
        const int kbase = kt * 16;

        // ---- logits tile: Q K^T (K loaded as B with same lane mapping) ----
        v8f cl = {};
#pragma unroll
        for (int c = 0; c < 8; ++c) {
            v2f bk = *(const v2f*)(km + (size_t)(kbase + lm) * D_SINGLE +
                                   h * HEAD_DIM + c * 4 + klo);
            cl = wmma4(aq[c], bk, cl);
        }

        const bool mk = mask[kbase + lm] != 0;   // this lane's key column
        float lg[8], pv[8], corr[8];
#pragma unroll
        for (int r = 0; r < 8; ++r) {
            const int row = r + 8 * half;
            float x = cl[r] + bias[((size_t)(qbase + row) * SEQ_L + kbase + lm) * N_HEADS + h];
            lg[r] = mk ? x : -1e30f;
        }

        // ---- online softmax (per-row over 16 lanes of each half) ----
#pragma unroll
        for (int r = 0; r < 8; ++r) {
            const int row = r + 8 * half;
            float rmax = lg[r];
            for (int off = 8; off >= 1; off >>= 1)
                rmax = fmaxf(rmax, __shfl_xor(rmax, off, 16));
            const float mo = Lm[row];
            const float mn = fmaxf(mo, rmax);
            const float pr = __expf(lg[r] - mn);
            float rsum = pr;
            for (int off = 8; off >= 1; off >>= 1)
                rsum += __shfl_xor(rsum, off, 16);
            const float cr = __expf(mo - mn);
            pv[r] = pr;
            corr[r] = cr;
            __builtin_amdgcn_wave_barrier();
            if (lm == 0) { Lm[row] = mn; Ll[row] = Ll[row] * cr + rsum; }
        }

        // ---- rescale acc, publish P tile (D layout -> LDS row-major) ----
#pragma unroll
        for (int r = 0; r < 8; ++r) {
            acc0[r] *= corr[r];
            acc1[r] *= corr[r];
            const int row = r + 8 * half;
            P[row * 16 + lm] = pv[r];
        }
        __builtin_amdgcn_wave_barrier();

        // ---- acc += P @ V  (A = P from LDS in A-layout, B = V tile) ----
#pragma unroll
        for (int c = 0; c < 4; ++c) {
            v2f ap = *(const v2f*)(P + lm * 16 + c * 4 + klo);
            const int kk = kbase + c * 4 + klo;
            const float* vrow0 = vm + (size_t)kk * D_SINGLE + h * HEAD_DIM;
            const float* vrow1 = vm + (size_t)(kk + 1) * D_SINGLE + h * HEAD_DIM;
            v2f b0, b1;
            b0.x = vrow0[lm];      b0.y = vrow1[lm];
            b1.x = vrow0[16 + lm]; b1.y = vrow1[16 + lm];
            acc0 = wmma4(ap, b0, acc0);
            acc1 = wmma4(ap, b1, acc1);
        }
        __builtin_amdgcn_wave_barrier();
    }

    // ---- epilogue: divide by l, write attended[q][h*32+d] ----
#pragma unroll
    for (int r = 0; r < 8; ++r) {
        const int row = r + 8 * half;
        const float linv = 1.0f / Ll[row];
        float* o = att + (size_t)(qbase + row) * D_SINGLE + h * HEAD_DIM;
        o[lm]      = acc0[r] * linv;
        o[16 + lm] = acc1[r] * linv;
    }
}

// ---------------------------------------------------------------------------
extern "C" void kernel_launch(void* const* d_in, const int* in_sizes, int n_in,
                              void* d_out, int out_size, void* d_ws, size_t ws_size,
                              hipStream_t stream) {
    const float* s    = (const float*)d_in[0];
    const float* pair = (const float*)d_in[1];
    const unsigned char* mask = (const unsigned char*)d_in[2];
    const float* g_s  = (const float*)d_in[3];
    const float* b_s  = (const float*)d_in[4];
    const float* Wq   = (const float*)d_in[5];
    const float* Wk   = (const float*)d_in[6];
    const float* Wv   = (const float*)d_in[7];
    const float* Wb   = (const float*)d_in[8];
    const float* Wo   = (const float*)d_in[9];
    const float* bo   = (const float*)d_in[10];
    const float* g_f  = (const float*)d_in[11];
    const float* b_f  = (const float*)d_in[12];
    const float* W1   = (const float*)d_in[13];
    const float* b1   = (const float*)d_in[14];
    const float* W2   = (const float*)d_in[15];
    const float* b2   = (const float*)d_in[16];
    float* out = (float*)d_out;

    const size_t NSD = (size_t)SEQ_L * D_SINGLE;          // 393216
    float* ws      = (float*)d_ws;
    float* s_norm  = ws;
    float* q       = s_norm + NSD;
    float* k       = q + NSD;
    float* v       = k + NSD;
    float* att     = v + NSD;
    float* s_res   = att + NSD;
    float* h_norm  = s_res + NSD;
    float* h1      = h_norm + NSD;                        // 1024 x 1536
    float* biasb   = h1 + (size_t)SEQ_L * 4 * D_SINGLE;   // 1024 x 1024 x 12

    // 1) s_norm = LN(s)
    ln_kernel<<<SEQ_L, 128, 0, stream>>>(s, g_s, b_s, s_norm);

    // 2) Q, K, V projections (1024x384 @ 384x384)
    gemm_kernel<<<dim3(6, 64), 128, 0, stream>>>(s_norm, Wq, nullptr, nullptr, q, SEQ_L, 384, 384, 0);
    gemm_kernel<<<dim3(6, 64), 128, 0, stream>>>(s_norm, Wk, nullptr, nullptr, k, SEQ_L, 384, 384, 0);
    gemm_kernel<<<dim3(6, 64), 128, 0, stream>>>(s_norm, Wv, nullptr, nullptr, v, SEQ_L, 384, 384, 0);

    // 3) pair bias: 1024 q-rows x 64 k-tiles = 65536 units, 4 units/block
    pair_bias_kernel<<<16384, 128, 0, stream>>>(pair, Wb, biasb);

    // 4) flash attention: 12 heads x 64 q-tiles = 768 units, 4 units/block
    attn_kernel<<<192, 128, 0, stream>>>(q, k, v, biasb, mask, att);

    // 5) s_res = s + att @ Wo + bo
    gemm_kernel<<<dim3(6, 64), 128, 0, stream>>>(att, Wo, bo, s, s_res, SEQ_L, 384, 384, 0);

    // 6) FFN: h_norm = LN(s_res); h1 = gelu(h_norm@W1 + b1); out = s_res + h1@W2 + b2
    ln_kernel<<<SEQ_L, 128, 0, stream>>>(s_res, g_f, b_f, h_norm);
    gemm_kernel<<<dim3(24, 64), 128, 0, stream>>>(h_norm, W1, b1, nullptr, h1, SEQ_L, 1536, 384, 1);
    gemm_kernel<<<dim3(6, 64), 128, 0, stream>>>(h1, W2, b2, s_res, out, SEQ_L, 384, 1536, 0);
}